// UnsupervisedLoss_26139170963578
// MI455X (gfx1250) — compile-verified
//
#include <hip/hip_runtime.h>
#include <hip/hip_bf16.h>

// ---------------------------------------------------------------------------
// Problem constants (from reference setup_inputs)
// ---------------------------------------------------------------------------
#define NB    8        // batch
#define NP    19248    // priors
#define NMD   32       // mask dim
#define NPH   138
#define NPW   138
#define NPIX  (NPH*NPW)   // 19044
#define NH    512
#define NW    512
#define NMID  64
#define NK    200      // TOP_K
#define NGS   16       // GAUSS_SAMPLES
#define NGD   64       // IOU_TRAIN_DIM
#define FEPS      1e-6f
#define CONF_T    0.05f
#define NMS_T     0.5f
#define MAXDET_N  100

// WMMA fp32 fragment types (V_WMMA_F32_16X16X4_F32: A/B = 2 VGPR, C/D = 8 VGPR)
typedef __attribute__((ext_vector_type(2))) float v2f;
typedef __attribute__((ext_vector_type(8))) float v8f;

// ---------------------------------------------------------------------------
// Workspace layout (float/int32 element offsets into d_ws)
// ---------------------------------------------------------------------------
constexpr size_t F_SCORE = 0;                               // NB*NP
constexpr size_t F_TOPI  = F_SCORE + (size_t)NB*NP;         // NB*NK (int)
constexpr size_t F_TOPV  = F_TOPI  + (size_t)NB*NK;         // NB*NK
constexpr size_t F_LOCK  = F_TOPV  + (size_t)NB*NK;         // NB*NK*4
constexpr size_t F_MASKK = F_LOCK  + (size_t)NB*NK*4;       // NB*NK*NMD
constexpr size_t F_IOU   = F_MASKK + (size_t)NB*NK*NMD;     // NB*NK*NK
constexpr size_t F_KEEPC = F_IOU   + (size_t)NB*NK*NK;      // NB*NK
constexpr size_t F_LOCC  = F_KEEPC + (size_t)NB*NK;         // NB*NK*4
constexpr size_t F_CONFC = F_LOCC  + (size_t)NB*NK*4;       // NB*NK
constexpr size_t F_MASKC = F_CONFC + (size_t)NB*NK;         // NB*NK*NMD
constexpr size_t F_PREDG = F_MASKC + (size_t)NB*NK*NMD;     // NB*NGS*NGS
constexpr size_t F_LOCG  = F_PREDG + (size_t)NB*NGS*NGS;    // NB*NGS*4
constexpr size_t F_VALG  = F_LOCG  + (size_t)NB*NGS*4;      // NB*NGS
constexpr size_t F_GG    = F_VALG  + (size_t)NB*NGS;        // NB*NGS*NGD*NGD
constexpr size_t F_FINAL = F_GG    + (size_t)NB*NGS*NGD*NGD;// NB*NPIX
constexpr size_t F_RESZ  = F_FINAL + (size_t)NB*NPIX;       // NB*NH*NW
// total ~3.4M elements (~13.5 MB)

// ---------------------------------------------------------------------------
// K0: init outputs
// ---------------------------------------------------------------------------
__global__ void k_init(float* out) {
  if (threadIdx.x == 0 && blockIdx.x == 0) { out[0] = 0.f; out[1] = 0.f; }
}

// ---------------------------------------------------------------------------
// K1: per-batch top-K via tournament selection.
//  - one full scan builds 256 per-thread (val,idx) maxima in LDS
//  - each of the 200 rounds: 8-step LDS reduction picks the winner
//    (value desc, index asc -- matches jax.lax.top_k), then ONLY the winning
//    thread rescans its ~76-element stride chunk.
// ---------------------------------------------------------------------------
__global__ void k_topk(const float* __restrict__ conf, const float* __restrict__ loc,
                       const float* __restrict__ mask, float* __restrict__ score,
                       int* __restrict__ topi, float* __restrict__ topv,
                       float* __restrict__ lock, float* __restrict__ maskk)
{
  const int b = blockIdx.x, tid = threadIdx.x;       // 256 threads
  __shared__ float smax[256];
  __shared__ int   simx[256];
  __shared__ float rval[256];
  __shared__ int   ridx[256];
  float* sc = score + (size_t)b*NP;

  for (int p = tid; p < NP; p += 256) {
    float c1 = conf[((size_t)b*NP + p)*2 + 1];
    sc[p] = (c1 > CONF_T) ? c1 : -1.0f;
  }
  __syncthreads();
  // initial per-thread max (ascending p => strict '>' keeps smallest index)
  {
    float bv = -3.f; int bi = NP;
    for (int p = tid; p < NP; p += 256) {
      float v = sc[p];
      if (v > bv) { bv = v; bi = p; }
    }
    smax[tid] = bv; simx[tid] = bi;
  }
  __syncthreads();

  for (int it = 0; it < NK; ++it) {
    rval[tid] = smax[tid]; ridx[tid] = simx[tid];
    __syncthreads();
    for (int off = 128; off > 0; off >>= 1) {
      if (tid < off) {
        float ov = rval[tid+off]; int oi = ridx[tid+off];
        if (ov > rval[tid] || (ov == rval[tid] && oi < ridx[tid])) {
          rval[tid] = ov; ridx[tid] = oi;
        }
      }
      __syncthreads();
    }
    const int   wi = ridx[0];
    const float wv = rval[0];
    if (tid == 0) { topi[b*NK+it] = wi; topv[b*NK+it] = wv; }
    if (tid == (wi & 255)) {            // owner thread (chunks are stride-256)
      sc[wi] = -2.0f;                   // mark taken
      float bv = -3.f; int bi = NP;
      for (int p = tid; p < NP; p += 256) {
        float v = sc[p];
        if (v > bv) { bv = v; bi = p; }
      }
      smax[tid] = bv; simx[tid] = bi;
    }
    __syncthreads();
  }
  // gather loc_k / mask_k
  for (int t = tid; t < NK*4; t += 256) {
    int k = t >> 2, d = t & 3;
    lock[(b*NK+k)*4 + d] = loc[((size_t)b*NP + topi[b*NK+k])*4 + d];
  }
  for (int t = tid; t < NK*NMD; t += 256) {
    int k = t / NMD, d = t % NMD;
    maskk[(b*NK+k)*NMD + d] = mask[((size_t)b*NP + topi[b*NK+k])*NMD + d];
  }
}

// ---------------------------------------------------------------------------
// K2: pairwise IoU MLP via fp32 WMMA.  One wave per 16 consecutive (i,j)
// pairs.  GEMM1 (M=16,K=8,N=64) = 2 WMMA K-steps x 4 N-tiles, fused with
// ReLU + W2 dot + sigmoid.  A fragments are loaded directly as aligned
// 64-bit vectors (khalf is an address offset, not a register select).
// ---------------------------------------------------------------------------
__global__ void k_iou_mlp(const float* __restrict__ lock, const float* __restrict__ W1,
                          const float* __restrict__ b1,   const float* __restrict__ W2,
                          const float* __restrict__ b2,   float* __restrict__ iou)
{
  const int gid  = blockIdx.x * blockDim.x + threadIdx.x;
  const int wave = gid >> 5;
  const int lane = threadIdx.x & 31;
  const int TILES = (NK*NK)/16;           // 2500
  if (wave >= NB*TILES) return;           // wave-uniform exit (EXEC stays full)
  const int b    = wave / TILES;
  const int tile = wave % TILES;
  const int base = tile * 16;

  // A fragment: lane<16 -> row m=lane, K-slots {0,1}; lane>=16 -> row m=lane-16, K-slots {2,3}
  const int m     = lane & 15;
  const int pair  = base + m;
  const int pi    = pair / NK;
  const int pj    = pair % NK;
  const int khalf = (lane < 16) ? 0 : 2;

  // grid row = [loc_k[pj](4) , loc_k[pi](4)]; K-step 0 = feats 0..3, step 1 = feats 4..7
  const v2f a0 = *(const v2f*)(lock + (size_t)(b*NK + pj)*4 + khalf);
  const v2f a1 = *(const v2f*)(lock + (size_t)(b*NK + pi)*4 + khalf);

  float acc[8];
  #pragma unroll
  for (int r = 0; r < 8; ++r) acc[r] = 0.f;

  const int ncol = lane & 15;
  #pragma unroll
  for (int nt = 0; nt < 4; ++nt) {
    const int n = nt*16 + ncol;
    v2f bf0, bf1;
    bf0.x = W1[(0 + khalf)*NMID + n];  bf0.y = W1[(1 + khalf)*NMID + n];
    bf1.x = W1[(4 + khalf)*NMID + n];  bf1.y = W1[(5 + khalf)*NMID + n];
    v8f c = {};
    c = __builtin_amdgcn_wmma_f32_16x16x4_f32(false, a0, false, bf0, (short)0, c, false, false);
    c = __builtin_amdgcn_wmma_f32_16x16x4_f32(false, a1, false, bf1, (short)0, c, false, false);
    const float w2n = W2[n];
    const float b1n = b1[n];
    #pragma unroll
    for (int r = 0; r < 8; ++r) {
      float h = c[r] + b1n;
      h = h > 0.f ? h : 0.f;                 // relu
      acc[r] += h * w2n;                     // fused hid @ W2
    }
  }
  // reduce the N dimension across the 16 lanes of each half
  #pragma unroll
  for (int r = 0; r < 8; ++r) {
    float v = acc[r];
    for (int off = 1; off < 16; off <<= 1) v += __shfl_xor(v, off, 32);
    acc[r] = v;
  }
  const float bb2 = b2[0];
  if (ncol == 0) {                          // lanes 0 (rows 0..7) and 16 (rows 8..15)
    const int rb = (lane < 16) ? 0 : 8;
    #pragma unroll
    for (int r = 0; r < 8; ++r) {
      const int pr = base + rb + r;
      iou[(size_t)b*NK*NK + pr] = 1.0f / (1.0f + __expf(-(acc[r] + bb2)));
    }
  }
}

// ---------------------------------------------------------------------------
// K3: NMS + cumsum cap + stable compaction; emit compacted loc/conf/mask/keep
// and the GAUSS_SAMPLES side outputs.  One block per batch.
// ---------------------------------------------------------------------------
__global__ void k_nms(const float* __restrict__ iou, const float* __restrict__ topv,
                      const float* __restrict__ lock, const float* __restrict__ maskk,
                      float* keepc, float* locc, float* confc, float* maskc,
                      float* predG, float* locG, float* valG)
{
  const int b = blockIdx.x, tid = threadIdx.x, NT = blockDim.x;
  __shared__ unsigned char keep[NK];
  __shared__ unsigned char valid[NK];
  __shared__ short perm[NK];
  for (int j = tid; j < NK; j += NT) valid[j] = topv[b*NK+j] > CONF_T;
  __syncthreads();
  for (int j = tid; j < NK; j += NT) {
    float mx = 0.f;
    if (valid[j])
      for (int i = 0; i < j; ++i)
        if (valid[i]) mx = fmaxf(mx, iou[(size_t)b*NK*NK + i*NK + j]);
    keep[j] = valid[j] && (mx <= NMS_T);
  }
  __syncthreads();
  if (tid == 0) {
    int cnt = 0;
    for (int j = 0; j < NK; ++j) if (keep[j]) { if (++cnt > MAXDET_N) keep[j] = 0; }
    int pos = 0;
    for (int j = 0; j < NK; ++j) if (keep[j])  perm[pos++] = (short)j;
    for (int j = 0; j < NK; ++j) if (!keep[j]) perm[pos++] = (short)j;
  }
  __syncthreads();
  for (int k = tid; k < NK; k += NT) {
    int p = perm[k];
    float kc = keep[p] ? 1.f : 0.f;
    keepc[b*NK+k] = kc;
    confc[b*NK+k] = kc * fmaxf(topv[b*NK+p], 0.f);
    for (int d = 0; d < 4; ++d) locc[(b*NK+k)*4+d] = lock[(b*NK+p)*4+d];
  }
  for (int t = tid; t < NK*NMD; t += NT) {
    int k = t / NMD, d = t % NMD;
    int p = perm[k];
    maskc[(b*NK+k)*NMD+d] = (keep[p] ? 1.f : 0.f) * maskk[(b*NK+p)*NMD+d];
  }
  for (int t = tid; t < NGS*NGS; t += NT) {
    int i = t / NGS, j = t % NGS;
    predG[b*NGS*NGS + t] = iou[(size_t)b*NK*NK + perm[i]*NK + perm[j]];
  }
  for (int s = tid; s < NGS; s += NT) {
    int p = perm[s];
    valG[b*NGS+s] = keep[p] ? 1.f : 0.f;
    for (int d = 0; d < 4; ++d) locG[(b*NGS+s)*4+d] = lock[(b*NK+p)*4+d];
  }
}

// ---------------------------------------------------------------------------
// K4: 64x64 gaussians for the 16 sampled detections per batch
// ---------------------------------------------------------------------------
__global__ void k_gaussG(const float* __restrict__ locG, float* __restrict__ gG)
{
  const int bs = blockIdx.x;            // NB*NGS blocks
  const float cx = locG[bs*4+0], cy = locG[bs*4+1];
  const float bw = locG[bs*4+2], bh = locG[bs*4+3];
  const float ivx = 1.f/(bw*0.5f + FEPS), ivy = 1.f/(bh*0.5f + FEPS);
  for (int t = threadIdx.x; t < NGD*NGD; t += blockDim.x) {
    int y = t / NGD, x = t % NGD;
    float xs = (x + 0.5f) * (1.f/NGD), ys = (y + 0.5f) * (1.f/NGD);
    float dx = (xs - cx) * ivx, dy = (ys - cy) * ivy;
    gG[(size_t)bs*NGD*NGD + t] = __expf(-0.5f*(dx*dx + dy*dy));
  }
}

// ---------------------------------------------------------------------------
// K5: pairwise gaussian IoU loss -> out[1].  One block per batch,
// thread t = pair (i,j).
// ---------------------------------------------------------------------------
__global__ void k_iouloss(const float* __restrict__ gG, const float* __restrict__ predG,
                          const float* __restrict__ valG, float* out)
{
  const int b = blockIdx.x, t = threadIdx.x;
  const int i = t / NGS, j = t % NGS;
  const float* gi = gG + ((size_t)b*NGS + i)*NGD*NGD;
  const float* gj = gG + ((size_t)b*NGS + j)*NGD*NGD;
  float inter = 0.f, uni = 0.f;
  for (int p = 0; p < NGD*NGD; ++p) {
    float a = gi[p], c = gj[p];
    inter += fminf(a, c);
    uni   += fmaxf(a, c);
  }
  float giou = inter / (uni + FEPS);
  float vm = valG[b*NGS+i] * valG[b*NGS+j];
  float d  = predG[b*NGS*NGS + t] - giou;
  __shared__ float sm[256];
  sm[t] = vm * d * d;
  __syncthreads();
  for (int off = 128; off > 0; off >>= 1) {
    if (t < off) sm[t] += sm[t+off];
    __syncthreads();
  }
  if (t == 0) atomicAdd(&out[1], sm[0]);
}

// ---------------------------------------------------------------------------
// K6: proto x mask_c GEMM via fp32 WMMA, fused with sigmoid, analytic
// gaussian, conf weighting and the per-pixel sum / sum-of-squares -> final.
// One wave per 16-pixel tile; loop over 13 detection tiles (K=32 -> 8 WMMAs).
// Next tile's B rows are prefetched (global_prefetch_b8) while WMMAs run.
// ---------------------------------------------------------------------------
__global__ void k_final_map(const float* __restrict__ proto, const float* __restrict__ maskc,
                            const float* __restrict__ locc,  const float* __restrict__ confc,
                            float* __restrict__ finalm)
{
  const int gid  = blockIdx.x * blockDim.x + threadIdx.x;
  const int wave = gid >> 5;
  const int lane = threadIdx.x & 31;
  const int TILES = (NPIX + 15)/16;       // 1191
  if (wave >= NB*TILES) return;           // wave-uniform
  const int b     = wave / TILES;
  const int ptile = wave % TILES;

  const int m     = lane & 15;
  const int khalf = (lane < 16) ? 0 : 2;
  int prow = ptile*16 + m;
  if (prow >= NPIX) prow = NPIX-1;        // pad rows read valid memory; outputs discarded
  const float* Arow = proto + ((size_t)b*NPIX + prow)*NMD;
  v2f afrag[8];
  #pragma unroll
  for (int ks = 0; ks < 8; ++ks)
    afrag[ks] = *(const v2f*)(Arow + ks*4 + khalf);

  float S[8], Q[8];
  #pragma unroll
  for (int r = 0; r < 8; ++r) { S[r] = 0.f; Q[r] = 0.f; }

  for (int nt = 0; nt < 13; ++nt) {       // 13*16 = 208 >= 200 detections
    const int n  = nt*16 + m;
    const int nc = (n < NK) ? n : 0;
    const float cf = (n < NK) ? confc[b*NK + nc] : 0.f;
    const float cx = locc[(b*NK+nc)*4+0], cy = locc[(b*NK+nc)*4+1];
    const float bw = locc[(b*NK+nc)*4+2], bh = locc[(b*NK+nc)*4+3];
    const float ivx = 1.f/(bw*0.5f + FEPS), ivy = 1.f/(bh*0.5f + FEPS);
    const float* Brow = maskc + ((size_t)b*NK + nc)*NMD;   // B[k][n] = mask_c[n][k]
    // prefetch next tile's B row into cache while this tile's WMMAs execute
    if (nt + 1 < 13) {
      const int nn = nt*16 + 16 + m;
      const int nnc = (nn < NK) ? nn : 0;
      __builtin_prefetch(maskc + ((size_t)b*NK + nnc)*NMD + khalf, 0, 0);
    }
    v8f c = {};
    #pragma unroll
    for (int ks = 0; ks < 8; ++ks) {
      const v2f bf = *(const v2f*)(Brow + ks*4 + khalf);
      c = __builtin_amdgcn_wmma_f32_16x16x4_f32(false, afrag[ks], false, bf, (short)0, c, false, false);
    }
    #pragma unroll
    for (int r = 0; r < 8; ++r) {
      const int pm = ptile*16 + ((lane < 16) ? r : 8 + r);
      if (pm < NPIX) {
        const int py = pm / NPW, px = pm % NPW;
        const float xs = (px + 0.5f) * (1.f/NPW);
        const float ys = (py + 0.5f) * (1.f/NPH);
        const float dx = (xs - cx) * ivx, dy = (ys - cy) * ivy;
        const float g  = __expf(-0.5f*(dx*dx + dy*dy));
        const float as = 1.0f / (1.0f + __expf(-c[r]));   // sigmoid(assembled)
        const float mv = as * g * cf;                     // mconf (keep folded into conf)
        S[r] += mv; Q[r] += mv*mv;
      }
    }
  }
  // reduce across detections held by the 16 lanes of each half
  #pragma unroll
  for (int r = 0; r < 8; ++r) {
    float s = S[r], q = Q[r];
    for (int off = 1; off < 16; off <<= 1) {
      s += __shfl_xor(s, off, 32);
      q += __shfl_xor(q, off, 32);
    }
    S[r] = s; Q[r] = q;
  }
  if ((lane & 15) == 0) {
    const int rb = (lane < 16) ? 0 : 8;
    #pragma unroll
    for (int r = 0; r < 8; ++r) {
      const int pm = ptile*16 + rb + r;
      if (pm < NPIX)
        finalm[(size_t)b*NPIX + pm] = 1.0f - Q[r] / (S[r] + FEPS);
    }
  }
}

// ---------------------------------------------------------------------------
// K7: bilinear resize (half-pixel centers, edge clamp) 138x138 -> 512x512
// ---------------------------------------------------------------------------
__global__ void k_resize(const float* __restrict__ finalm, float* __restrict__ resized)
{
  const int idx = blockIdx.x * blockDim.x + threadIdx.x;
  if (idx >= NB*NH*NW) return;
  const int b = idx / (NH*NW);
  const int hw = idx % (NH*NW);
  const int h = hw / NW, w = hw % NW;
  const float fy = (h + 0.5f) * ((float)NPH/NH) - 0.5f;
  const float fx = (w + 0.5f) * ((float)NPW/NW) - 0.5f;
  const int y0 = (int)floorf(fy), x0 = (int)floorf(fx);
  const float wy = fy - (float)y0, wx = fx - (float)x0;
  const int y0c = min(max(y0, 0), NPH-1), y1c = min(max(y0+1, 0), NPH-1);
  const int x0c = min(max(x0, 0), NPW-1), x1c = min(max(x0+1, 0), NPW-1);
  const float* f = finalm + (size_t)b*NPIX;
  const float v00 = f[y0c*NPW+x0c], v01 = f[y0c*NPW+x1c];
  const float v10 = f[y1c*NPW+x0c], v11 = f[y1c*NPW+x1c];
  resized[idx] = (1.f-wy)*((1.f-wx)*v00 + wx*v01) + wy*((1.f-wx)*v10 + wx*v11);
}

// ---------------------------------------------------------------------------
// K8: weighted variance reduction -> out[0]
// var = sum_{b,c,p} (orig - wmean[c,p])^2 * r[b,p] / (total[p]+eps)
// ---------------------------------------------------------------------------
__global__ void k_var(const float* __restrict__ original, const float* __restrict__ resized,
                      float* out)
{
  const int idx = blockIdx.x * blockDim.x + threadIdx.x;   // pixel index
  float local = 0.f;
  if (idx < NH*NW) {
    float r[NB]; float total = 0.f;
    #pragma unroll
    for (int b = 0; b < NB; ++b) { r[b] = resized[(size_t)b*NH*NW + idx]; total += r[b]; }
    const float inv = 1.0f / (total + FEPS);
    #pragma unroll
    for (int ch = 0; ch < 3; ++ch) {
      float o[NB]; float wm = 0.f;
      #pragma unroll
      for (int b = 0; b < NB; ++b) {
        o[b] = original[((size_t)b*3 + ch)*NH*NW + idx];
        wm += o[b]*r[b];
      }
      #pragma unroll
      for (int b = 0; b < NB; ++b) {
        float d = o[b] - wm;
        local += d*d*r[b]*inv;
      }
    }
  }
  __shared__ float sm[256];
  sm[threadIdx.x] = local;
  __syncthreads();
  for (int off = 128; off > 0; off >>= 1) {
    if (threadIdx.x < off) sm[threadIdx.x] += sm[threadIdx.x+off];
    __syncthreads();
  }
  if (threadIdx.x == 0) atomicAdd(&out[0], sm[0]);
}

// ---------------------------------------------------------------------------
// Launch
// ---------------------------------------------------------------------------
extern "C" void kernel_launch(void* const* d_in, const int* in_sizes, int n_in,
                              void* d_out, int out_size, void* d_ws, size_t ws_size,
                              hipStream_t stream) {
  (void)in_sizes; (void)n_in; (void)out_size; (void)ws_size;
  const float* original = (const float*)d_in[0];
  const float* loc      = (const float*)d_in[1];
  const float* conf     = (const float*)d_in[2];
  const float* mask     = (const float*)d_in[3];
  const float* proto    = (const float*)d_in[4];
  const float* W1       = (const float*)d_in[5];
  const float* b1       = (const float*)d_in[6];
  const float* W2       = (const float*)d_in[7];
  const float* b2       = (const float*)d_in[8];
  float* out = (float*)d_out;
  float* ws  = (float*)d_ws;

  float* s_score = ws + F_SCORE;
  int*   s_topi  = (int*)(ws + F_TOPI);
  float* s_topv  = ws + F_TOPV;
  float* s_lock  = ws + F_LOCK;
  float* s_maskk = ws + F_MASKK;
  float* s_iou   = ws + F_IOU;
  float* s_keepc = ws + F_KEEPC;
  float* s_locc  = ws + F_LOCC;
  float* s_confc = ws + F_CONFC;
  float* s_maskc = ws + F_MASKC;
  float* s_predG = ws + F_PREDG;
  float* s_locG  = ws + F_LOCG;
  float* s_valG  = ws + F_VALG;
  float* s_gG    = ws + F_GG;
  float* s_final = ws + F_FINAL;
  float* s_resz  = ws + F_RESZ;

  k_init<<<1, 64, 0, stream>>>(out);

  k_topk<<<NB, 256, 0, stream>>>(conf, loc, mask, s_score, s_topi, s_topv, s_lock, s_maskk);

  {
    const int waves = NB * (NK*NK/16);                 // 20000 waves
    const int threads = waves * 32;
    k_iou_mlp<<<(threads + 127)/128, 128, 0, stream>>>(s_lock, W1, b1, W2, b2, s_iou);
  }

  k_nms<<<NB, 256, 0, stream>>>(s_iou, s_topv, s_lock, s_maskk,
                                s_keepc, s_locc, s_confc, s_maskc,
                                s_predG, s_locG, s_valG);

  k_gaussG<<<NB*NGS, 256, 0, stream>>>(s_locG, s_gG);
  k_iouloss<<<NB, 256, 0, stream>>>(s_gG, s_predG, s_valG, out);

  {
    const int waves = NB * ((NPIX + 15)/16);           // 9528 waves
    const int threads = waves * 32;
    k_final_map<<<(threads + 127)/128, 128, 0, stream>>>(proto, s_maskc, s_locc, s_confc, s_final);
  }

  k_resize<<<(NB*NH*NW + 255)/256, 256, 0, stream>>>(s_final, s_resz);
  k_var<<<(NH*NW + 255)/256, 256, 0, stream>>>(original, s_resz, out);

  (void)s_keepc;
}